// Seq2Seq_10453950398998
// MI455X (gfx1250) — compile-verified
//
#include <hip/hip_runtime.h>
#include <math.h>

#define S_LEN 128
#define B_SZ  128
#define C_IN  256
#define E_SZ  256
#define D_SZ  512
#define EMB_SZ 256
#define V_SZ 10000
#define T_SZ 32

typedef __attribute__((ext_vector_type(2))) float v2f;
typedef __attribute__((ext_vector_type(8))) float v8f;

// ---------------------------------------------------------------------------
// Wave-per-tile WMMA f32 GEMM with addend/activation:
//   C[M,N] = act( A[M,K] @ W[N,K]^T + bias[N] + addend )
// grid.x = N/16, grid.y = M/16, blockDim.x = 32. K % 4 == 0.
// Used for the latency-bound recurrent steps and small decoder GEMMs where
// wave-level parallelism matters more than operand reuse.
// ---------------------------------------------------------------------------
__global__ void wmma_gemm_f32(const float* __restrict__ A, int lda,
                              const float* __restrict__ W, int ldw,
                              const float* __restrict__ bias,
                              const float* __restrict__ addend, int ldadd,
                              float* __restrict__ Cout, int ldc,
                              int K, int act)
{
    const int lane = threadIdx.x;
    const int r16  = lane & 15;
    const int half = lane >> 4;           // 0 or 1
    const int n0 = blockIdx.x * 16;
    const int m0 = blockIdx.y * 16;

    const float* Arow = A + (size_t)(m0 + r16) * (size_t)lda;
    const float* Wrow = W + (size_t)(n0 + r16) * (size_t)ldw;

    v8f acc = {0.f, 0.f, 0.f, 0.f, 0.f, 0.f, 0.f, 0.f};
    for (int k = 0; k < K; k += 4) {
        const int kk = k + half * 2;
        v2f a, b;
        a.x = Arow[kk];     a.y = Arow[kk + 1];
        b.x = Wrow[kk];     b.y = Wrow[kk + 1];
        acc = __builtin_amdgcn_wmma_f32_16x16x4_f32(
            false, a, false, b, (short)0, acc, false, false);
    }

    const int col = n0 + r16;
    const float bv = bias ? bias[col] : 0.0f;
#pragma unroll
    for (int i = 0; i < 8; ++i) {
        const int rg = m0 + i + half * 8;
        float v = acc[i] + bv;
        if (addend) v += addend[(size_t)rg * (size_t)ldadd + col];
        if (act == 1) v = tanhf(v);
        Cout[(size_t)rg * (size_t)ldc + col] = v;
    }
}

// ---------------------------------------------------------------------------
// Register-blocked wave-tile WMMA GEMM: each wave computes a (16*MU)x(16*NU)
// tile; per K-step loads MU A-frags + NU B-frags and issues MU*NU WMMAs.
// grid.x = N/(16*NU), grid.y = M/(16*MU), blockDim.x = 32.
// Used for the compute-heavy GEMMs (input pre-projections, enc_part,
// vocab projection) to raise the WMMA : vmem ratio and cut L2 traffic.
// ---------------------------------------------------------------------------
template <int MU, int NU>
__global__ void wmma_gemm_tiled(const float* __restrict__ A, int lda,
                                const float* __restrict__ W, int ldw,
                                const float* __restrict__ bias,
                                float* __restrict__ Cout, int ldc,
                                int K, int act)
{
    const int lane = threadIdx.x;
    const int r16  = lane & 15;
    const int half = lane >> 4;
    const int n0 = blockIdx.x * (16 * NU);
    const int m0 = blockIdx.y * (16 * MU);

    const float* Arow[MU];
    const float* Wrow[NU];
#pragma unroll
    for (int i = 0; i < MU; ++i)
        Arow[i] = A + (size_t)(m0 + 16 * i + r16) * (size_t)lda;
#pragma unroll
    for (int j = 0; j < NU; ++j)
        Wrow[j] = W + (size_t)(n0 + 16 * j + r16) * (size_t)ldw;

    v8f acc[MU][NU];
#pragma unroll
    for (int i = 0; i < MU; ++i)
#pragma unroll
        for (int j = 0; j < NU; ++j)
            acc[i][j] = (v8f){0.f, 0.f, 0.f, 0.f, 0.f, 0.f, 0.f, 0.f};

    for (int k = 0; k < K; k += 4) {
        const int kk = k + half * 2;
        v2f a[MU], b[NU];
#pragma unroll
        for (int i = 0; i < MU; ++i) { a[i].x = Arow[i][kk]; a[i].y = Arow[i][kk + 1]; }
#pragma unroll
        for (int j = 0; j < NU; ++j) { b[j].x = Wrow[j][kk]; b[j].y = Wrow[j][kk + 1]; }
#pragma unroll
        for (int i = 0; i < MU; ++i)
#pragma unroll
            for (int j = 0; j < NU; ++j)
                acc[i][j] = __builtin_amdgcn_wmma_f32_16x16x4_f32(
                    false, a[i], false, b[j], (short)0, acc[i][j], false, false);
    }

#pragma unroll
    for (int j = 0; j < NU; ++j) {
        const int col = n0 + 16 * j + r16;
        const float bv = bias ? bias[col] : 0.0f;
#pragma unroll
        for (int i = 0; i < MU; ++i) {
#pragma unroll
            for (int q = 0; q < 8; ++q) {
                const int rg = m0 + 16 * i + q + half * 8;
                float v = acc[i][j][q] + bv;
                if (act == 1) v = tanhf(v);
                Cout[(size_t)rg * (size_t)ldc + col] = v;
            }
        }
    }
}

// ---------------------------------------------------------------------------
// scores[s*B+b] = sum_d tanh(hW[b,d] + enc_part[s*B+b, d]) * attn_v[d]
// ---------------------------------------------------------------------------
__global__ void attn_scores_k(const float* __restrict__ hW,
                              const float* __restrict__ enc_part,
                              const float* __restrict__ attn_v,
                              float* __restrict__ scores)
{
    const int wave = threadIdx.x >> 5;
    const int lane = threadIdx.x & 31;
    const int idx  = blockIdx.x * 8 + wave;      // 0 .. S*B-1
    const int b    = idx & (B_SZ - 1);
    const float* ep = enc_part + (size_t)idx * D_SZ;
    const float* hb = hW + (size_t)b * D_SZ;
    float acc = 0.f;
    for (int d = lane; d < D_SZ; d += 32)
        acc += tanhf(hb[d] + ep[d]) * attn_v[d];
    for (int o = 16; o > 0; o >>= 1)
        acc += __shfl_down(acc, o, 32);
    if (lane == 0) scores[idx] = acc;
}

// softmax over S for each batch column b (scores layout [s*B + b])
__global__ void softmax_s_k(const float* __restrict__ scores, float* __restrict__ a)
{
    const int b = threadIdx.x;                  // 128 threads
    float m = -3.402823466e38f;
    for (int s = 0; s < S_LEN; ++s) m = fmaxf(m, scores[s * B_SZ + b]);
    float sum = 0.f;
    for (int s = 0; s < S_LEN; ++s) {
        float e = expf(scores[s * B_SZ + b] - m);
        a[s * B_SZ + b] = e;
        sum += e;
    }
    const float inv = 1.0f / sum;
    for (int s = 0; s < S_LEN; ++s) a[s * B_SZ + b] *= inv;
}

// weighted[b,e] = sum_s a[s,b] * enc_out[s,b,e]; scatter into xbuf & predin
__global__ void weighted_k(const float* __restrict__ a,
                           const float* __restrict__ enc_out,
                           float* __restrict__ xbuf, float* __restrict__ predin)
{
    const int b = blockIdx.x;
    const int e = threadIdx.x;                  // 512
    float acc = 0.f;
    for (int s = 0; s < S_LEN; ++s)
        acc += a[s * B_SZ + b] * enc_out[((size_t)(s * B_SZ + b)) * (2 * E_SZ) + e];
    xbuf[b * (EMB_SZ + 2 * E_SZ) + EMB_SZ + e] = acc;
    predin[b * (D_SZ + 2 * E_SZ + EMB_SZ) + D_SZ + e] = acc;
}

// embedding gather; scatter into xbuf cols [0,256) and predin cols [1024,1280)
__global__ void embed_k(const int* __restrict__ trg, int t,
                        const float* __restrict__ emb,
                        float* __restrict__ xbuf, float* __restrict__ predin)
{
    const int b = blockIdx.x;
    const int e = threadIdx.x;                  // 256
    const int tok = trg[b * T_SZ + t];
    const float v = emb[(size_t)tok * EMB_SZ + e];
    xbuf[b * (EMB_SZ + 2 * E_SZ) + e] = v;
    predin[b * (D_SZ + 2 * E_SZ + EMB_SZ) + (D_SZ + 2 * E_SZ) + e] = v;
}

// GRU gate combine; updates hidden in place, copies h_new into predin[:,0:512)
__global__ void gru_k(const float* __restrict__ gi, const float* __restrict__ gh,
                      float* __restrict__ hidden, float* __restrict__ predin)
{
    const int b = blockIdx.x;
    const int d = threadIdx.x;                  // 512
    const float* gib = gi + (size_t)b * 3 * D_SZ;
    const float* ghb = gh + (size_t)b * 3 * D_SZ;
    const float r = 1.0f / (1.0f + expf(-(gib[d] + ghb[d])));
    const float z = 1.0f / (1.0f + expf(-(gib[D_SZ + d] + ghb[D_SZ + d])));
    const float n = tanhf(gib[2 * D_SZ + d] + r * ghb[2 * D_SZ + d]);
    const float h = hidden[b * D_SZ + d];
    const float hn = (1.0f - z) * n + z * h;
    hidden[b * D_SZ + d] = hn;
    predin[b * (D_SZ + 2 * E_SZ + EMB_SZ) + d] = hn;
}

// hcat[b,:] = concat(h_f[b], h_b[b]) from the interleaved enc_out buffer
__global__ void hcat_k(const float* __restrict__ enc_out, float* __restrict__ hcat)
{
    const int b = blockIdx.x;
    const int j = threadIdx.x;                  // 512
    const float v = (j < E_SZ)
        ? enc_out[((size_t)((S_LEN - 1) * B_SZ + b)) * 512 + j]  // h_f (t=S-1, cols 0:256)
        : enc_out[((size_t)b) * 512 + j];                        // h_b (t=0,  cols 256:512)
    hcat[b * 512 + j] = v;
}

// softmax over V=10000 in place on d_out row (b, t)
__global__ void softmax_v_k(float* __restrict__ out, int t)
{
    __shared__ float red[256];
    const int b = blockIdx.x;
    const int tid = threadIdx.x;
    float* row = out + (size_t)b * T_SZ * V_SZ + (size_t)t * V_SZ;
    float m = -3.402823466e38f;
    for (int i = tid; i < V_SZ; i += 256) m = fmaxf(m, row[i]);
    red[tid] = m; __syncthreads();
    for (int o = 128; o > 0; o >>= 1) {
        if (tid < o) red[tid] = fmaxf(red[tid], red[tid + o]);
        __syncthreads();
    }
    m = red[0]; __syncthreads();
    float sum = 0.f;
    for (int i = tid; i < V_SZ; i += 256) {
        const float e = expf(row[i] - m);
        row[i] = e;
        sum += e;
    }
    red[tid] = sum; __syncthreads();
    for (int o = 128; o > 0; o >>= 1) {
        if (tid < o) red[tid] += red[tid + o];
        __syncthreads();
    }
    const float inv = 1.0f / red[0];
    for (int i = tid; i < V_SZ; i += 256) row[i] *= inv;
}

__global__ void zero_k(float* __restrict__ p, int n)
{
    const int i = blockIdx.x * 256 + threadIdx.x;
    if (i < n) p[i] = 0.0f;
}

// ---------------------------------------------------------------------------
extern "C" void kernel_launch(void* const* d_in, const int* in_sizes, int n_in,
                              void* d_out, int out_size, void* d_ws, size_t ws_size,
                              hipStream_t stream)
{
    (void)in_sizes; (void)n_in; (void)out_size; (void)ws_size;

    const float* src    = (const float*)d_in[0];
    const int*   trg    = (const int*)  d_in[1];
    const float* Wih_f  = (const float*)d_in[2];
    const float* Whh_f  = (const float*)d_in[3];
    const float* bih_f  = (const float*)d_in[4];
    const float* bhh_f  = (const float*)d_in[5];
    const float* Wih_b  = (const float*)d_in[6];
    const float* Whh_b  = (const float*)d_in[7];
    const float* bih_b  = (const float*)d_in[8];
    const float* bhh_b  = (const float*)d_in[9];
    const float* fcW    = (const float*)d_in[10];
    const float* fcb    = (const float*)d_in[11];
    const float* attn_W = (const float*)d_in[12];
    const float* attn_b = (const float*)d_in[13];
    const float* attn_v = (const float*)d_in[14];
    const float* emb    = (const float*)d_in[15];
    const float* gWih   = (const float*)d_in[16];
    const float* gWhh   = (const float*)d_in[17];
    const float* gbih   = (const float*)d_in[18];
    const float* gbhh   = (const float*)d_in[19];
    const float* outW   = (const float*)d_in[20];
    const float* outb   = (const float*)d_in[21];
    float* out = (float*)d_out;

    float* ws = (float*)d_ws;
    size_t off = 0;
    float* pre_f   = ws + off; off += (size_t)S_LEN * B_SZ * E_SZ;       // 4M f
    float* pre_b   = ws + off; off += (size_t)S_LEN * B_SZ * E_SZ;       // 4M f
    float* enc_part = pre_f;  // reuse pre_f+pre_b region after scans (same size)
    float* enc_out = ws + off; off += (size_t)S_LEN * B_SZ * 2 * E_SZ;   // 8M f
    float* hcat    = ws + off; off += (size_t)B_SZ * 512;
    float* hidden  = ws + off; off += (size_t)B_SZ * D_SZ;
    float* hW      = ws + off; off += (size_t)B_SZ * D_SZ;
    float* zbuf    = ws + off; off += (size_t)B_SZ * E_SZ;
    float* scores  = ws + off; off += (size_t)S_LEN * B_SZ;
    float* attw    = ws + off; off += (size_t)S_LEN * B_SZ;
    float* xbuf    = ws + off; off += (size_t)B_SZ * (EMB_SZ + 2 * E_SZ);
    float* predin  = ws + off; off += (size_t)B_SZ * (D_SZ + 2 * E_SZ + EMB_SZ);
    float* gi      = ws + off; off += (size_t)B_SZ * 3 * D_SZ;
    float* gh      = ws + off; off += (size_t)B_SZ * 3 * D_SZ;

    const dim3 wblk(32);

    // h0 = 0
    zero_k<<<(B_SZ * E_SZ + 255) / 256, 256, 0, stream>>>(zbuf, B_SZ * E_SZ);

    // pre_f / pre_b : (S*B, C) @ (E, C)^T + bih    [big: 4x4 wave tiles]
    wmma_gemm_tiled<4, 4><<<dim3(E_SZ / 64, (S_LEN * B_SZ) / 64), wblk, 0, stream>>>(
        src, C_IN, Wih_f, C_IN, bih_f, pre_f, E_SZ, C_IN, 0);
    wmma_gemm_tiled<4, 4><<<dim3(E_SZ / 64, (S_LEN * B_SZ) / 64), wblk, 0, stream>>>(
        src, C_IN, Wih_b, C_IN, bih_b, pre_b, E_SZ, C_IN, 0);

    // forward scan: h_t = tanh(pre_f[t] + h_{t-1} @ Whh_f^T + bhh_f) -> enc_out[t, :, 0:256]
    for (int t = 0; t < S_LEN; ++t) {
        const float* Aprev = (t == 0) ? zbuf : (enc_out + (size_t)(t - 1) * B_SZ * 512);
        const int lda = (t == 0) ? E_SZ : 512;
        wmma_gemm_f32<<<dim3(E_SZ / 16, B_SZ / 16), wblk, 0, stream>>>(
            Aprev, lda, Whh_f, E_SZ, bhh_f,
            pre_f + (size_t)t * B_SZ * E_SZ, E_SZ,
            enc_out + (size_t)t * B_SZ * 512, 512, E_SZ, 1);
    }
    // backward scan -> enc_out[t, :, 256:512]
    for (int i = 0; i < S_LEN; ++i) {
        const int t = S_LEN - 1 - i;
        const float* Aprev = (i == 0) ? zbuf
                                      : (enc_out + (size_t)(t + 1) * B_SZ * 512 + E_SZ);
        const int lda = (i == 0) ? E_SZ : 512;
        wmma_gemm_f32<<<dim3(E_SZ / 16, B_SZ / 16), wblk, 0, stream>>>(
            Aprev, lda, Whh_b, E_SZ, bhh_b,
            pre_b + (size_t)t * B_SZ * E_SZ, E_SZ,
            enc_out + (size_t)t * B_SZ * 512 + E_SZ, 512, E_SZ, 1);
    }

    // hidden = tanh(concat(h_f, h_b) @ fcW^T + fcb)
    hcat_k<<<B_SZ, 512, 0, stream>>>(enc_out, hcat);
    wmma_gemm_f32<<<dim3(D_SZ / 16, B_SZ / 16), wblk, 0, stream>>>(
        hcat, 512, fcW, 512, fcb, nullptr, 0, hidden, D_SZ, 512, 1);

    // enc_part = enc_out @ W_e^T + attn_b  (W_e = attn_W[:, D:], row stride 1024)
    wmma_gemm_tiled<4, 4><<<dim3(D_SZ / 64, (S_LEN * B_SZ) / 64), wblk, 0, stream>>>(
        enc_out, 2 * E_SZ, attn_W + D_SZ, D_SZ + 2 * E_SZ, attn_b,
        enc_part, D_SZ, 2 * E_SZ, 0);

    // ---- decoder ----
    for (int t = 0; t < T_SZ; ++t) {
        embed_k<<<B_SZ, EMB_SZ, 0, stream>>>(trg, t, emb, xbuf, predin);

        // hW = hidden @ W_h^T  (W_h = attn_W[:, :D])
        wmma_gemm_f32<<<dim3(D_SZ / 16, B_SZ / 16), wblk, 0, stream>>>(
            hidden, D_SZ, attn_W, D_SZ + 2 * E_SZ, nullptr, nullptr, 0,
            hW, D_SZ, D_SZ, 0);

        attn_scores_k<<<(S_LEN * B_SZ) / 8, 256, 0, stream>>>(hW, enc_part, attn_v, scores);
        softmax_s_k<<<1, B_SZ, 0, stream>>>(scores, attw);
        weighted_k<<<B_SZ, 2 * E_SZ, 0, stream>>>(attw, enc_out, xbuf, predin);

        // gi = x @ gru_Wih^T + bih ; gh = h @ gru_Whh^T + bhh
        wmma_gemm_f32<<<dim3((3 * D_SZ) / 16, B_SZ / 16), wblk, 0, stream>>>(
            xbuf, EMB_SZ + 2 * E_SZ, gWih, EMB_SZ + 2 * E_SZ, gbih, nullptr, 0,
            gi, 3 * D_SZ, EMB_SZ + 2 * E_SZ, 0);
        wmma_gemm_f32<<<dim3((3 * D_SZ) / 16, B_SZ / 16), wblk, 0, stream>>>(
            hidden, D_SZ, gWhh, D_SZ, gbhh, nullptr, 0,
            gh, 3 * D_SZ, D_SZ, 0);

        gru_k<<<B_SZ, D_SZ, 0, stream>>>(gi, gh, hidden, predin);

        // vocab projection -> d_out[b, t, :] directly (ldc = T*V), 2x5 wave tiles
        // (N = 10000 = 125 * 80, M = 128 = 4 * 32)
        wmma_gemm_tiled<2, 5><<<dim3(V_SZ / 80, B_SZ / 32), wblk, 0, stream>>>(
            predin, D_SZ + 2 * E_SZ + EMB_SZ, outW, D_SZ + 2 * E_SZ + EMB_SZ,
            outb, out + (size_t)t * V_SZ, T_SZ * V_SZ,
            D_SZ + 2 * E_SZ + EMB_SZ, 0);
        softmax_v_k<<<B_SZ, 256, 0, stream>>>(out, t);
    }
}